// TensorProductLinearGate_32615981646157
// MI455X (gfx1250) — compile-verified
//
#include <hip/hip_runtime.h>

// ---------------------------------------------------------------------------
// TensorProductLinearGate for MI455X (gfx1250), f32 WMMA 16x16x4 path.
//   N = 65536 rows, MUL = 256, K = 512.
//   s   = tp0 @ W0 * 1/sqrt(512) + b0   (512 cols: 256 scalars + 256 gates)
//   v_i = tp1_i @ W1 * 1/sqrt(512)      (256 cols, i = 0..2)
//   out[n] = [ gelu(s[:256]) | (v * sigmoid(gates))[c,i] at 256+3c+i ]
// tp matrices are synthesized on the fly from raw x rows staged in LDS.
// B fragments are double-buffered (prefetch k+4 before WMMAs of k).
// ---------------------------------------------------------------------------

#define ROWS_PER_BLK 32
#define S_STRIDE     1028   // 1024 x-floats + 4 pad -> bank=(4*row+c)%64, conflict-free A frags

typedef float v2f __attribute__((ext_vector_type(2)));
typedef float v4f __attribute__((ext_vector_type(4)));
typedef float v8f __attribute__((ext_vector_type(8)));

__device__ __forceinline__ v8f wmma_f32(v2f a, v2f b, v8f c) {
  // V_WMMA_F32_16X16X4_F32 : D(16x16 f32) = A(16x4) * B(4x16) + C
  return __builtin_amdgcn_wmma_f32_16x16x4_f32(false, a, false, b, (short)0, c,
                                               false, false);
}

__device__ __forceinline__ float fast_sigmoid(float t) {
  // 1/(1+e^-t) via v_exp_f32 + v_rcp_f32
  return __builtin_amdgcn_rcpf(1.0f + __expf(-t));
}

__global__ __launch_bounds__(256)
void tpg_wmma_kernel(const float* __restrict__ x, const float* __restrict__ y,
                     const float* __restrict__ W0, const float* __restrict__ b0,
                     const float* __restrict__ W1, float* __restrict__ out) {
  constexpr float kInvSqrt3   = 0.5773502691896258f;   // 1/sqrt(3)
  constexpr float kInvSqrtFan = 0.04419417382415922f;  // 1/sqrt(512)

  extern __shared__ float sx[];  // ROWS_PER_BLK rows of x, stride S_STRIDE floats
  const int tid      = threadIdx.x;
  const int row_base = blockIdx.x * ROWS_PER_BLK;

  // ---- Stage 32 rows of x into LDS (NT 128-bit loads; x read once from HBM) ----
  {
    const v4f* xg = reinterpret_cast<const v4f*>(x + (size_t)row_base * 1024);
    for (int f4 = tid; f4 < ROWS_PER_BLK * 256; f4 += 256) {
      const int row = f4 >> 8;    // 256 vec4 per row
      const int c4  = f4 & 255;
      v4f v = __builtin_nontemporal_load(&xg[row * 256 + c4]);
      *reinterpret_cast<v4f*>(&sx[row * S_STRIDE + c4 * 4]) = v;  // ds_store_b128
    }
  }
  __syncthreads();

  const int wave = tid >> 5;
  const int lane = tid & 31;
  const int half = lane >> 4;   // K-half select for A/B fragments
  const int mrow = lane & 15;   // A row within 16-tile / B,C column within tile
  const int koff = 2 * half;

  // Per-lane y scalars for both 16-row tiles handled by this wave.
  float y0r[2], y1r[2][3], y1s[2][3];  // y1s = y1 * 1/sqrt(3) (phase-2 tp0 coeffs)
  #pragma unroll
  for (int rt = 0; rt < 2; ++rt) {
    const float* yp = y + (size_t)(row_base + rt * 16 + mrow) * 4;
    y0r[rt] = yp[0];
    #pragma unroll
    for (int i = 0; i < 3; ++i) {
      y1r[rt][i] = yp[1 + i];
      y1s[rt][i] = yp[1 + i] * kInvSqrt3;
    }
  }

  // A-fragment LDS base pointers (koff folded in; affine in k after that).
  const float* spA = &sx[mrow * S_STRIDE + koff];            // phase 1: + k
  const float* spB = &sx[mrow * S_STRIDE + 256 + 3 * koff];  // phase 2: + 3*(k-256)

  // Each wave owns a 32-wide output-column strip, as two 16-col subtiles.
  for (int sub = 0; sub < 2; ++sub) {
    const int o_base = (wave * 2 + sub) * 16;
    const int col    = o_base + mrow;     // output column (0..255)

    v8f accS[2] = {}, accG[2] = {}, accV0[2] = {}, accV1[2] = {}, accV2[2] = {};

    // B fragment pointers; B row index is (k + koff) and (k + koff + 1).
    const float* w0p = W0 + (size_t)koff * 512 + col;   // [0]=bs.x [2048B]=bs.y [+256]=bg
    const float* w1p = W1 + (size_t)koff * 256 + col;

    auto loadB = [&](int k, v2f& bs, v2f& bg, v2f& bw) {
      const float* p0 = w0p + (size_t)k * 512;
      const float* p1 = w1p + (size_t)k * 256;
      bs.x = p0[0];   bs.y = p0[512];
      bg.x = p0[256]; bg.y = p0[768];
      bw.x = p1[0];   bw.y = p1[256];
    };

    auto step1 = [&](int k, v2f bs, v2f bg, v2f bw) {
      #pragma unroll
      for (int rt = 0; rt < 2; ++rt) {
        const float* sp = spA + rt * 16 * S_STRIDE + k;
        v2f xv;  xv.x = sp[0];  xv.y = sp[1];     // ds_load_2addr_b32
        v2f a0, a1, a2, a3;
        a0.x = xv.x * y0r[rt];     a0.y = xv.y * y0r[rt];      // tp0
        a1.x = xv.x * y1r[rt][0];  a1.y = xv.y * y1r[rt][0];   // tp1_0
        a2.x = xv.x * y1r[rt][1];  a2.y = xv.y * y1r[rt][1];   // tp1_1
        a3.x = xv.x * y1r[rt][2];  a3.y = xv.y * y1r[rt][2];   // tp1_2
        accS [rt] = wmma_f32(a0, bs, accS [rt]);
        accG [rt] = wmma_f32(a0, bg, accG [rt]);
        accV0[rt] = wmma_f32(a1, bw, accV0[rt]);
        accV1[rt] = wmma_f32(a2, bw, accV1[rt]);
        accV2[rt] = wmma_f32(a3, bw, accV2[rt]);
      }
    };

    auto step2 = [&](int k, v2f bs, v2f bg, v2f bw) {
      #pragma unroll
      for (int rt = 0; rt < 2; ++rt) {
        const float* sp = spB + rt * 16 * S_STRIDE + 3 * (k - 256);
        const float e00 = sp[0], e01 = sp[1], e02 = sp[2];   // x1[c',   0..2]
        const float e10 = sp[3], e11 = sp[4], e12 = sp[5];   // x1[c'+1, 0..2]
        v2f a0, a1, a2, a3;
        a0.x = fmaf(e00, y1s[rt][0], fmaf(e01, y1s[rt][1], e02 * y1s[rt][2]));
        a0.y = fmaf(e10, y1s[rt][0], fmaf(e11, y1s[rt][1], e12 * y1s[rt][2]));
        a1.x = e00 * y0r[rt];  a1.y = e10 * y0r[rt];
        a2.x = e01 * y0r[rt];  a2.y = e11 * y0r[rt];
        a3.x = e02 * y0r[rt];  a3.y = e12 * y0r[rt];
        accS [rt] = wmma_f32(a0, bs, accS [rt]);
        accG [rt] = wmma_f32(a0, bg, accG [rt]);
        accV0[rt] = wmma_f32(a1, bw, accV0[rt]);
        accV1[rt] = wmma_f32(a2, bw, accV1[rt]);
        accV2[rt] = wmma_f32(a3, bw, accV2[rt]);
      }
    };

    // ---- Software-pipelined K sweep: prefetch B(k+4) before WMMAs of k ----
    v2f bs, bg, bw, bsN, bgN, bwN;
    loadB(0, bs, bg, bw);
    for (int k = 0; k < 256; k += 4) {          // phase 1: tp cols from x0
      loadB(k + 4, bsN, bgN, bwN);              // k+4 <= 256: rows <= 259 valid
      step1(k, bs, bg, bw);
      bs = bsN; bg = bgN; bw = bwN;
    }
    // bs/bg/bw now hold B(k=256): first phase-2 fragment.
    for (int k = 256; k < 508; k += 4) {        // phase 2: tp cols from x1 triples
      loadB(k + 4, bsN, bgN, bwN);              // k+4 <= 508: rows <= 511 valid
      step2(k, bs, bg, bw);
      bs = bsN; bg = bgN; bw = bwN;
    }
    step2(508, bs, bg, bw);                     // peeled: no OOB prefetch

    // ---- Epilogue: scale, bias, gelu via sigmoid(2t), gating, NT stores ----
    const float bs0 = b0[col];
    const float bg0 = b0[col + 256];
    #pragma unroll
    for (int rt = 0; rt < 2; ++rt) {
      #pragma unroll
      for (int r = 0; r < 8; ++r) {
        const int    m = r + 8 * half;                 // C layout: VGPR r, lane half
        const size_t n = (size_t)(row_base + rt * 16 + m);
        const float s = accS[rt][r] * kInvSqrtFan + bs0;
        const float g = accG[rt][r] * kInvSqrtFan + bg0;
        // gelu(s) = s * sigmoid(1.595769...*s + 0.0713548...*s^3)   (tanh form)
        const float s3  = s * s * s;
        const float gel = s * fast_sigmoid(fmaf(1.5957691216057308f, s,
                                                0.07135481627f * s3));
        const float sig = fast_sigmoid(g);
        float* op = out + n * 1024;
        __builtin_nontemporal_store(gel, &op[col]);
        __builtin_nontemporal_store(accV0[rt][r] * kInvSqrtFan * sig, &op[256 + 3 * col + 0]);
        __builtin_nontemporal_store(accV1[rt][r] * kInvSqrtFan * sig, &op[256 + 3 * col + 1]);
        __builtin_nontemporal_store(accV2[rt][r] * kInvSqrtFan * sig, &op[256 + 3 * col + 2]);
      }
    }
  }
}

extern "C" void kernel_launch(void* const* d_in, const int* in_sizes, int n_in,
                              void* d_out, int out_size, void* d_ws, size_t ws_size,
                              hipStream_t stream) {
  const float* x  = (const float*)d_in[0];   // (N, 1024)
  const float* y  = (const float*)d_in[1];   // (N, 4)
  const float* W0 = (const float*)d_in[2];   // (512, 512)
  const float* b0 = (const float*)d_in[3];   // (512,)
  const float* W1 = (const float*)d_in[4];   // (512, 256)
  float* out = (float*)d_out;                // (N, 1024)

  const int nrows = in_sizes[0] / 1024;      // 65536
  const dim3 grid(nrows / ROWS_PER_BLK);
  const dim3 block(256);
  const size_t lds_bytes = (size_t)ROWS_PER_BLK * S_STRIDE * sizeof(float); // ~128.5 KB

  hipLaunchKernelGGL(tpg_wmma_kernel, grid, block, lds_bytes, stream,
                     x, y, W0, b0, W1, out);
}